// Classifier_62311385530651
// MI455X (gfx1250) — compile-verified
//
#include <hip/hip_runtime.h>

// MI455X / gfx1250 implementation.
// Math: dist(q,c) = u^T (A G A - 2A) u + ||q||^2, u = q C^T, A = (G+lam I)^-1.
// ||q||^2 cancels under per-row min-max, so only the n=20-dim form matters.
// Dominant cost: U = X C^T  (3200x512 @ 512x1280 f32 GEMM) -> V_WMMA_F32_16X16X4_F32.

typedef __attribute__((ext_vector_type(2))) float v2f;
typedef __attribute__((ext_vector_type(8))) float v8f;

#define C_CLASSES 64
#define N_SUP     20
#define D_DIM     512
#define B_BATCH   128
#define RES       25
#define Q_TOT     (B_BATCH * RES)        // 3200
#define LAMBDA    (20.0f / 512.0f)

// ---------------------------------------------------------------------------
// Kernel 1: per class c, compute S_c = A G A - 2A  (20x20), A=(G+lam I)^-1.
// One block per class, 256 threads. Gauss-Jordan (SPD, no pivoting needed).
// ---------------------------------------------------------------------------
__global__ __launch_bounds__(256) void prep_kernel(const float* __restrict__ high,
                                                   float* __restrict__ S_out) {
  __shared__ float Ch[N_SUP * D_DIM];      // 40 KB: class support rows
  __shared__ float G[N_SUP * N_SUP];
  __shared__ float Aug[N_SUP * 40];        // [G+lam I | I]
  __shared__ float Ainv[N_SUP * N_SUP];
  __shared__ float T[N_SUP * N_SUP];
  __shared__ float fsh[N_SUP];

  const int c = blockIdx.x;
  const int tid = threadIdx.x;

  for (int i = tid; i < N_SUP * D_DIM; i += 256)
    Ch[i] = high[(size_t)c * N_SUP * D_DIM + i];
  __syncthreads();

  // Gram matrix G = C C^T
  for (int e = tid; e < N_SUP * N_SUP; e += 256) {
    const int i = e / N_SUP, j = e % N_SUP;
    const float* ri = &Ch[i * D_DIM];
    const float* rj = &Ch[j * D_DIM];
    float s = 0.f;
    for (int k = 0; k < D_DIM; ++k) s += ri[k] * rj[k];
    G[e] = s;
    Aug[i * 40 + j]      = s + (i == j ? LAMBDA : 0.f);
    Aug[i * 40 + 20 + j] = (i == j) ? 1.f : 0.f;
  }
  __syncthreads();

  // Gauss-Jordan elimination (no row normalization; divide at the end)
  for (int k = 0; k < N_SUP; ++k) {
    if (tid < N_SUP)
      fsh[tid] = (tid == k) ? 0.f : Aug[tid * 40 + k] / Aug[k * 40 + k];
    __syncthreads();
    for (int e = tid; e < N_SUP * 40; e += 256) {
      const int r = e / 40, j = e % 40;
      if (r != k) Aug[e] -= fsh[r] * Aug[k * 40 + j];
    }
    __syncthreads();
  }
  for (int e = tid; e < N_SUP * N_SUP; e += 256) {
    const int i = e / N_SUP, j = e % N_SUP;
    Ainv[e] = Aug[i * 40 + 20 + j] / Aug[i * 40 + i];
  }
  __syncthreads();

  // T = G * Ainv
  for (int e = tid; e < N_SUP * N_SUP; e += 256) {
    const int i = e / N_SUP, j = e % N_SUP;
    float s = 0.f;
    for (int k = 0; k < N_SUP; ++k) s += G[i * N_SUP + k] * Ainv[k * N_SUP + j];
    T[e] = s;
  }
  __syncthreads();

  // S = Ainv * T - 2*Ainv
  for (int e = tid; e < N_SUP * N_SUP; e += 256) {
    const int i = e / N_SUP, j = e % N_SUP;
    float s = 0.f;
    for (int k = 0; k < N_SUP; ++k) s += Ainv[i * N_SUP + k] * T[k * N_SUP + j];
    S_out[(size_t)c * N_SUP * N_SUP + e] = s - 2.f * Ainv[e];
  }
}

// ---------------------------------------------------------------------------
// Kernel 2: U[c][q][0..19] = sum_d X[q][d] * high[c][n][d]  via f32 WMMA.
// Grid (25, 64); block 256 = 8 waves. Wave w: rows q0..q0+15, cols 0..31
// (cols 20..31 are zero pad), K-loop over d in steps of 4.
// A frag (16x4 f32): lane<16 -> M=lane, K={0,1}; lane>=16 -> K={2,3}.
// B frag (4x16 f32): lane<16 -> N=lane, K={0,1}; lane>=16 -> K={2,3}.
// ---------------------------------------------------------------------------
__global__ __launch_bounds__(256) void u_gemm_kernel(const float* __restrict__ x,
                                                     const float* __restrict__ high,
                                                     float* __restrict__ U) {
  __shared__ float Bh[32 * D_DIM];         // 64 KB, rows 20..31 zeroed
  const int c = blockIdx.y;
  const int tid = threadIdx.x;
  const int lane = tid & 31;
  const int wv = tid >> 5;

  for (int i = tid; i < N_SUP * D_DIM; i += 256)
    Bh[i] = high[(size_t)c * N_SUP * D_DIM + i];
  for (int i = tid; i < (32 - N_SUP) * D_DIM; i += 256)
    Bh[N_SUP * D_DIM + i] = 0.f;
  __syncthreads();

  const int q0   = blockIdx.x * 128 + wv * 16;
  const int nlo  = lane & 15;
  const int koff = (lane >> 4) * 2;
  const float* Arow = &x[(size_t)(q0 + nlo) * D_DIM];

  v8f acc0 = {};
  v8f acc1 = {};
  for (int k = 0; k < D_DIM; k += 4) {
    v2f a  = *(const v2f*)&Arow[k + koff];
    v2f b0 = *(const v2f*)&Bh[nlo * D_DIM + k + koff];
    v2f b1 = *(const v2f*)&Bh[(16 + nlo) * D_DIM + k + koff];
    acc0 = __builtin_amdgcn_wmma_f32_16x16x4_f32(false, a, false, b0,
                                                 (short)0, acc0, false, false);
    acc1 = __builtin_amdgcn_wmma_f32_16x16x4_f32(false, a, false, b1,
                                                 (short)0, acc1, false, false);
  }

  // C/D layout: VGPR r, lanes 0-15 -> M=r; lanes 16-31 -> M=8+r; N=lane%16.
  const int rbase = q0 + 8 * (lane >> 4);
  for (int r = 0; r < 8; ++r) {
    const int qq = rbase + r;
    U[((size_t)c * Q_TOT + qq) * N_SUP + nlo] = acc0[r];
  }
  if (nlo < 4) {
    for (int r = 0; r < 8; ++r) {
      const int qq = rbase + r;
      U[((size_t)c * Q_TOT + qq) * N_SUP + 16 + nlo] = acc1[r];
    }
  }
}

// ---------------------------------------------------------------------------
// Kernel 3: per batch element b: L[c] = -(1/25) * sum_res u^T S_c u,
// then per-row min-max. One block per b; fixed-order reductions (deterministic).
// ---------------------------------------------------------------------------
__global__ __launch_bounds__(256) void reduce_kernel(const float* __restrict__ U,
                                                     const float* __restrict__ S,
                                                     float* __restrict__ out) {
  __shared__ float Ssh[C_CLASSES * N_SUP * N_SUP];   // 102.4 KB
  __shared__ float part[256];
  __shared__ float L[C_CLASSES];
  __shared__ float mnmx[2];

  const int b = blockIdx.x;
  const int tid = threadIdx.x;

  for (int i = tid; i < C_CLASSES * N_SUP * N_SUP; i += 256) Ssh[i] = S[i];
  __syncthreads();

  const int c = tid >> 2;          // 64 classes
  const int g = tid & 3;           // 4 res groups
  const float* Sc = &Ssh[c * N_SUP * N_SUP];
  float acc = 0.f;
  for (int res = g; res < RES; res += 4) {
    const int q = b * RES + res;
    const float* u = &U[((size_t)c * Q_TOT + q) * N_SUP];
    float ur[N_SUP];
    for (int i = 0; i < N_SUP; ++i) ur[i] = u[i];
    float qf = 0.f;
    for (int i = 0; i < N_SUP; ++i) {
      float s = 0.f;
      for (int j = 0; j < N_SUP; ++j) s += Sc[i * N_SUP + j] * ur[j];
      qf += ur[i] * s;
    }
    acc += qf;
  }
  part[tid] = acc;
  __syncthreads();

  if (tid < C_CLASSES) {
    const float ssum = part[tid * 4] + part[tid * 4 + 1] +
                       part[tid * 4 + 2] + part[tid * 4 + 3];
    L[tid] = -ssum * (1.0f / RES);
  }
  __syncthreads();

  if (tid == 0) {
    float mn = L[0], mx = L[0];
    for (int i = 1; i < C_CLASSES; ++i) {
      mn = fminf(mn, L[i]);
      mx = fmaxf(mx, L[i]);
    }
    mnmx[0] = mn;
    mnmx[1] = mx;
  }
  __syncthreads();

  if (tid < C_CLASSES)
    out[b * C_CLASSES + tid] = (L[tid] - mnmx[0]) / (mnmx[1] - mnmx[0]);
}

// ---------------------------------------------------------------------------
extern "C" void kernel_launch(void* const* d_in, const int* in_sizes, int n_in,
                              void* d_out, int out_size, void* d_ws, size_t ws_size,
                              hipStream_t stream) {
  const float* x    = (const float*)d_in[0];   // (128, 25, 512)
  const float* high = (const float*)d_in[1];   // (64, 20, 512)
  float* out = (float*)d_out;                  // (128, 64)

  float* S = (float*)d_ws;                                     // 64*400 f32 = 102400 B
  float* U = (float*)((char*)d_ws + (size_t)C_CLASSES * 400 * sizeof(float)); // 16.4 MB

  prep_kernel<<<C_CLASSES, 256, 0, stream>>>(high, S);
  u_gemm_kernel<<<dim3(Q_TOT / 128, C_CLASSES), 256, 0, stream>>>(x, high, U);
  reduce_kernel<<<B_BATCH, 256, 0, stream>>>(U, S, out);
}